// MemoryBank_21973052686345
// MI455X (gfx1250) — compile-verified
//
#include <hip/hip_runtime.h>
#include <math.h>

typedef float v2f __attribute__((ext_vector_type(2)));
typedef float v8f __attribute__((ext_vector_type(8)));

#define C_NUM 4
#define Q_SZ 256
#define TOPK 8
#define DIM 64

// ws layout (floats): [0,256) meanW (c*64+d, already /Q); [256,260) sums; [260,264) counts

__global__ __launch_bounds__(256) void mb_prep(const float* __restrict__ memory,
                                               float* __restrict__ ws) {
    int t = threadIdx.x;                 // 256 threads: c = t/64, d = t%64
    int c = t >> 6, d = t & 63;
    float s = 0.f;
    const float* base = memory + (size_t)c * Q_SZ * DIM + d;
    for (int q = 0; q < Q_SZ; ++q) s += base[q * DIM];
    ws[t] = s * (1.0f / (float)Q_SZ);    // fold /Q (TEMPERATURE == 1)
    if (t < 2 * C_NUM) ws[256 + t] = 0.f;  // zero sums+counts every launch
}

__global__ __launch_bounds__(256) void mb_ce(const float* __restrict__ F,
                                             const int* __restrict__ label,
                                             const float* __restrict__ ws,
                                             float* __restrict__ accum /* = ws+256 */) {
    // Zero-padded B tile: 16 rows (classes 0..3 real, 4..15 zero) x 64 K.
    __shared__ float sB[16 * DIM];
    __shared__ float lsum[C_NUM];
    __shared__ float lcnt[C_NUM];
    int t = threadIdx.x;
    #pragma unroll
    for (int i = t; i < 16 * DIM; i += 256)
        sB[i] = (i < C_NUM * DIM) ? ws[i] : 0.f;
    if (t < C_NUM) { lsum[t] = 0.f; lcnt[t] = 0.f; }
    __syncthreads();

    int wave = t >> 5;
    int lane = t & 31;
    int half = lane >> 4;        // A: K pair {0,1} vs {2,3}; D: rows 0-7 vs 8-15
    int lrow = lane & 15;
    int rowBase = blockIdx.x * 128 + wave * 16;

    const v2f* Arow = (const v2f*)(F + (size_t)(rowBase + lrow) * DIM + 2 * half);
    const v2f* Bcol = (const v2f*)(sB + lrow * DIM + 2 * half);

    // Preload operands: 16 global b64 + 16 LDS b64, issued back-to-back.
    v2f a[16], b[16];
    #pragma unroll
    for (int kb = 0; kb < 16; ++kb) a[kb] = Arow[kb * 2];
    #pragma unroll
    for (int kb = 0; kb < 16; ++kb) b[kb] = Bcol[kb * 2];

    v8f acc = {};
    #pragma unroll
    for (int kb = 0; kb < 16; ++kb)
        acc = __builtin_amdgcn_wmma_f32_16x16x4_f32(false, a[kb], false, b[kb],
                                                    (short)0, acc, false, false);

    // D layout: vgpr v, lanes 0-15 -> (m = v,   n = lane)
    //                     lanes 16-31 -> (m = v+8, n = lane-16)
    // Classes live in n = 0..3 -> lanes {0..3} and {16..19}; quads closed under xor 1/2.
    if (lrow < C_NUM) {
        int cls = lrow;
        #pragma unroll
        for (int v = 0; v < 8; ++v) {
            int m = rowBase + v + 8 * half;
            float x = acc[v];
            float mx = fmaxf(x, __shfl_xor(x, 1, 32));
            mx = fmaxf(mx, __shfl_xor(mx, 2, 32));
            float e = expf(x - mx);
            float s = e + __shfl_xor(e, 1, 32);
            s = s + __shfl_xor(s, 2, 32);
            if (label[m] == cls) {
                float loss = -(x - mx - logf(s));
                atomicAdd(&lsum[cls], loss);
                atomicAdd(&lcnt[cls], 1.0f);
            }
        }
    }
    __syncthreads();
    if (t < C_NUM) {
        atomicAdd(&accum[t], lsum[t]);
        atomicAdd(&accum[C_NUM + t], lcnt[t]);
    }
}

__global__ void mb_finalize(const float* __restrict__ ws, float* __restrict__ out) {
    float total = 0.f;
    int present = 0;
    for (int c = 0; c < C_NUM; ++c) {
        float cnt = ws[260 + c];
        if (cnt > 0.f) { total += ws[256 + c] / cnt; ++present; }
    }
    out[0] = total / (float)(present > 0 ? present : 1);
}

__global__ __launch_bounds__(256) void mb_topk_update(const float* __restrict__ conf,
                                                      const int* __restrict__ label,
                                                      const float* __restrict__ F,
                                                      const float* __restrict__ memory,
                                                      float* __restrict__ outMem,
                                                      int N) {
    int c = blockIdx.x;          // one block per class
    int t = threadIdx.x;
    const float NEG = -INFINITY;

    // per-thread sorted top-8 (value desc, index asc on ties)
    float val[TOPK];
    int   idx[TOPK];
    #pragma unroll
    for (int i = 0; i < TOPK; ++i) { val[i] = NEG; idx[i] = 0x7fffffff; }

    for (int n = t; n < N; n += 256) {
        if (label[n] != c) continue;
        float v = conf[n];
        if (v > val[TOPK - 1] || (v == val[TOPK - 1] && n < idx[TOPK - 1])) {
            int p = TOPK - 1;
            while (p > 0 && (v > val[p - 1] || (v == val[p - 1] && n < idx[p - 1]))) {
                val[p] = val[p - 1]; idx[p] = idx[p - 1]; --p;
            }
            val[p] = v; idx[p] = n;
        }
    }

    __shared__ float cval[256];
    __shared__ int   cidx[256];
    __shared__ int   top_idx[TOPK];
    __shared__ int   s_nvalid;
    int nv = 0;

    for (int r = 0; r < TOPK; ++r) {
        cval[t] = val[0]; cidx[t] = idx[0];
        __syncthreads();
        for (int off = 128; off > 0; off >>= 1) {
            if (t < off) {
                float v2 = cval[t + off]; int i2 = cidx[t + off];
                if (v2 > cval[t] || (v2 == cval[t] && i2 < cidx[t])) {
                    cval[t] = v2; cidx[t] = i2;
                }
            }
            __syncthreads();
        }
        float wv = cval[0];
        int widx = cidx[0];
        if (t == 0) {
            top_idx[r] = widx;
            if (wv > NEG) ++nv;       // finite winner
        }
        // winner, if in this thread, is its local max (slot 0): pop it
        if (val[0] == wv && idx[0] == widx) {
            #pragma unroll
            for (int k = 0; k < TOPK - 1; ++k) { val[k] = val[k + 1]; idx[k] = idx[k + 1]; }
            val[TOPK - 1] = NEG; idx[TOPK - 1] = 0x7fffffff;
        }
        __syncthreads();
    }
    if (t == 0) s_nvalid = nv;
    __syncthreads();
    int nvalid = s_nvalid;

    const float* memC = memory + (size_t)c * Q_SZ * DIM;
    float* outC = outMem + (size_t)c * Q_SZ * DIM;
    for (int e = t; e < Q_SZ * DIM; e += 256) {
        int q = e >> 6;
        int d = e & 63;
        float x;
        if (q < nvalid) x = F[(size_t)top_idx[q] * DIM + d];
        else            x = memC[(size_t)(q - nvalid) * DIM + d];
        outC[e] = x;
    }
}

extern "C" void kernel_launch(void* const* d_in, const int* in_sizes, int n_in,
                              void* d_out, int out_size, void* d_ws, size_t ws_size,
                              hipStream_t stream) {
    const float* conf   = (const float*)d_in[0];
    const int*   label  = (const int*)d_in[1];
    const float* F      = (const float*)d_in[2];   // [N, 64]
    const float* memory = (const float*)d_in[3];   // [4, 256, 64]
    float* out = (float*)d_out;                    // [0]=loss, [1..65536]=new_memory
    float* ws  = (float*)d_ws;
    int N = in_sizes[0];

    mb_prep<<<1, 256, 0, stream>>>(memory, ws);
    mb_ce<<<N / 128, 256, 0, stream>>>(F, label, ws, ws + 256);
    mb_finalize<<<1, 1, 0, stream>>>(ws, out);
    mb_topk_update<<<C_NUM, 256, 0, stream>>>(conf, label, F, memory, out + 1, N);
}